// SoftNCutLoss3D_91070486545041
// MI455X (gfx1250) — compile-verified
//
#include <hip/hip_runtime.h>
#include <hip/hip_bf16.h>

// ---------------------------------------------------------------------------
// SoftNCutLoss3D, fully fused for gfx1250 (MI455X).
//   IMG=30, RADIUS=3, OI=11, OX=3 -> effective stencil is 7x3x3 = 63 taps
//   (weight kernel is zero outside |dc|<=1, |dm|<=1).
// One wave (wave32) owns 16 voxels; the 63-tap weighted reductions run as a
// GEMV on the matrix pipe: V_WMMA_F32_16X16X4_F32, K padded to 64, weight
// vector in B column 0.  Pg's v_exp_f32 (TRANS) co-executes with the WMMA
// (XDL) chain.  Inner loop is branch-free: unconditional loads at clamped
// (always-valid) addresses, zero-pad semantics applied with v_cndmask; all
// k-decode folds to compile-time constants selected by one per-lane bool.
// Deterministic: shfl-xor tree -> LDS fixed-order -> per-block partials in
// d_ws -> single-block fixed-order finalize.
// ---------------------------------------------------------------------------

#define IMG  30
#define NVOX (IMG * IMG * IMG)          /* 27000 */
#define WAVES_PER_BLOCK 8
#define BLOCK_THREADS   (WAVES_PER_BLOCK * 32)
#define VOX_PER_WAVE    16
#define VOX_PER_BLOCK   (WAVES_PER_BLOCK * VOX_PER_WAVE)   /* 128 */
#define NBLOCKS         ((NVOX + VOX_PER_BLOCK - 1) / VOX_PER_BLOCK) /* 211 */

typedef float v2f __attribute__((ext_vector_type(2)));
typedef float v8f __attribute__((ext_vector_type(8)));

#if defined(__has_builtin)
#if __has_builtin(__builtin_amdgcn_wmma_f32_16x16x4_f32)
#define USE_WMMA_F32 1
#endif
#endif

// k = b*9 + (dc+1)*3 + (dm+1); b in [0,7) -> db = b-3; dc,dm in {-1,0,1}.
// k == 63 is the K=64 zero pad.  All-constant when called with literal k
// (the kk/j loops are fully unrolled), so decode folds away entirely.
struct KDec { int db, dc, dm; float w; };
__device__ __forceinline__ constexpr KDec kdecode(int k) {
    if (k >= 63) return {0, 0, 0, 0.0f};
    const int b   = k / 9;
    const int rem = k - 9 * b;
    const int t3  = rem / 3;
    const int dc  = t3 - 1;
    const int dm  = (rem - 3 * t3) - 1;
    const int db  = b - 3;
    const int dw2 = dc * dc + 2 * dm * dm;           // in {0,1,2,3}
    const float w = (dw2 == 0) ? 7.0f                // 7*exp(-dw2/9)
                  : (dw2 == 1) ? 6.2638752f
                  : (dw2 == 2) ? 5.6051618f
                  :              5.0157192f;
    return {db, dc, dm, w};
}

__device__ __forceinline__ int iclamp(int v, int lo, int hi) {
    return v < lo ? lo : (v > hi ? hi : v);
}

__global__ __launch_bounds__(BLOCK_THREADS)
void softncut_partials(const float* __restrict__ img,
                       const float* __restrict__ enc,
                       float* __restrict__ ws)
{
    const int  lane = threadIdx.x & 31;
    const int  wave = threadIdx.x >> 5;
    const int  m    = lane & 15;               // voxel slot within wave tile
    const bool hiK  = (lane >= 16);            // A-matrix K split per ISA layout

    const int  n     = (blockIdx.x * WAVES_PER_BLOCK + wave) * VOX_PER_WAVE + m;
    const bool valid = (n < NVOX);
    const int  nc    = valid ? n : (NVOX - 1); // safe address for masked loads

    const int xi = nc / (IMG * IMG);
    const int r0 = nc - xi * IMG * IMG;
    const int yj = r0 / IMG;
    const int zl = r0 - yj * IMG;

    const float center = img[nc];              // unconditional; pg masked later
    const float e_raw  = enc[nc];
    const float e_n    = valid ? e_raw : 0.0f;

#if defined(USE_WMMA_F32)
    v8f acc_den = {0.f,0.f,0.f,0.f,0.f,0.f,0.f,0.f};
    v8f acc_n0  = {0.f,0.f,0.f,0.f,0.f,0.f,0.f,0.f};
    v8f acc_n1  = {0.f,0.f,0.f,0.f,0.f,0.f,0.f,0.f};
#else
    float p_den = 0.f, p_n0 = 0.f, p_n1 = 0.f;
#endif

#pragma unroll
    for (int kk = 0; kk < 16; ++kk) {
        float ag[2], a0g[2], a1g[2], wfull[2];
#pragma unroll
        for (int j = 0; j < 2; ++j) {
            // Both candidate taps are compile-time; pick with one bool.
            const KDec lo = kdecode(kk * 4 + 0 + j);   // lanes 0..15
            const KDec hi = kdecode(kk * 4 + 2 + j);   // lanes 16..31
            const int   db = hiK ? hi.db : lo.db;
            const int   dc = hiK ? hi.dc : lo.dc;
            const int   dm = hiK ? hi.dm : lo.dm;
            const float w  = hiK ? hi.w  : lo.w;       // 0 for the k=63 pad

            const int xx = xi + db, yy = yj + dc, zz = zl + dm;
            const bool inb = ((unsigned)xx < (unsigned)IMG) &
                             ((unsigned)yy < (unsigned)IMG) &
                             ((unsigned)zz < (unsigned)IMG);
            // Branch-free: UNCONDITIONAL loads at a clamped, always-valid
            // address; zero-pad semantics applied afterwards with cndmask.
            const int idx = (iclamp(xx, 0, IMG - 1) * IMG +
                             iclamp(yy, 0, IMG - 1)) * IMG +
                             iclamp(zz, 0, IMG - 1);
            const float pv_raw = img[idx];             // always executed
            const float ev_raw = enc[idx];             // single enc load
            const float pv = inb ? pv_raw : 0.0f;
            const float ev = inb ? ev_raw : 0.0f;
            const float s1 = inb ? (1.0f - ev_raw) : 0.0f;

            const float d  = pv - center;
            float pg = __expf(-d * d * (1.0f / 121.0f));   // exp(-(dx)^2/OI^2)
            if (!valid) pg = 0.0f;                         // dead-voxel rows

            ag[j]    = pg;
            a0g[j]   = pg * ev;
            a1g[j]   = pg * s1;
            wfull[j] = w;
        }
#if defined(USE_WMMA_F32)
        // B 4x16: weight vector in column N=0 only (lanes 0 and 16).
        v2f A  = { ag[0],  ag[1]  };
        v2f A0 = { a0g[0], a0g[1] };
        v2f A1 = { a1g[0], a1g[1] };
        v2f B  = { (m == 0) ? wfull[0] : 0.f, (m == 0) ? wfull[1] : 0.f };
        acc_den = __builtin_amdgcn_wmma_f32_16x16x4_f32(false, A,  false, B, (short)0, acc_den, false, false);
        acc_n0  = __builtin_amdgcn_wmma_f32_16x16x4_f32(false, A0, false, B, (short)0, acc_n0,  false, false);
        acc_n1  = __builtin_amdgcn_wmma_f32_16x16x4_f32(false, A1, false, B, (short)0, acc_n1,  false, false);
#else
        p_den += ag[0]  * wfull[0] + ag[1]  * wfull[1];
        p_n0  += a0g[0] * wfull[0] + a0g[1] * wfull[1];
        p_n1  += a1g[0] * wfull[0] + a1g[1] * wfull[1];
#endif
    }

    // Per-voxel field values into lane m (lanes 16..31 mirror).
    float den_m, n0_m, n1_m;
#if defined(USE_WMMA_F32)
    {
        // D column N=0: rows 0..7 in lane 0 VGPR r, rows 8..15 in lane 16 VGPR r.
        const int src = (m < 8) ? 0 : 16;
        float dsel = 0.f, s0sel = 0.f, s1sel = 0.f;
#pragma unroll
        for (int r = 0; r < 8; ++r) {
            const float td = __shfl(acc_den[r], src, 32);
            const float t0 = __shfl(acc_n0[r],  src, 32);
            const float t1 = __shfl(acc_n1[r],  src, 32);
            if ((m & 7) == r) { dsel = td; s0sel = t0; s1sel = t1; }
        }
        den_m = dsel; n0_m = s0sel; n1_m = s1sel;
    }
#else
    den_m = p_den + __shfl_xor(p_den, 16, 32);
    n0_m  = p_n0  + __shfl_xor(p_n0,  16, 32);
    n1_m  = p_n1  + __shfl_xor(p_n1,  16, 32);
#endif

    // Four global accumulators:
    //   cA = sum e*nom0, cB = sum e*den, cC = sum (1-e)*nom1, cD = sum (1-e)*den
    float cA = 0.f, cB = 0.f, cC = 0.f, cD = 0.f;
    if (lane < 16 && valid) {
        cA = e_n * n0_m;
        cB = e_n * den_m;
        cC = (1.0f - e_n) * n1_m;
        cD = (1.0f - e_n) * den_m;
    }
#pragma unroll
    for (int off = 16; off > 0; off >>= 1) {
        cA += __shfl_xor(cA, off, 32);
        cB += __shfl_xor(cB, off, 32);
        cC += __shfl_xor(cC, off, 32);
        cD += __shfl_xor(cD, off, 32);
    }

    __shared__ float red[WAVES_PER_BLOCK][4];
    if (lane == 0) {
        red[wave][0] = cA; red[wave][1] = cB; red[wave][2] = cC; red[wave][3] = cD;
    }
    __syncthreads();
    if (threadIdx.x == 0) {
        float A = 0.f, B = 0.f, C = 0.f, D = 0.f;
        for (int wv = 0; wv < WAVES_PER_BLOCK; ++wv) {   // fixed order: deterministic
            A += red[wv][0]; B += red[wv][1]; C += red[wv][2]; D += red[wv][3];
        }
        float* o = ws + (size_t)blockIdx.x * 4;
        o[0] = A; o[1] = B; o[2] = C; o[3] = D;
    }
}

__global__ void softncut_finalize(const float* __restrict__ ws,
                                  float* __restrict__ out)
{
    if (blockIdx.x == 0 && threadIdx.x == 0) {
        float A = 0.f, B = 0.f, C = 0.f, D = 0.f;
        for (int b = 0; b < NBLOCKS; ++b) {              // fixed order: deterministic
            A += ws[4 * b + 0];
            B += ws[4 * b + 1];
            C += ws[4 * b + 2];
            D += ws[4 * b + 3];
        }
        // loss = mean(K_CLUST - (r0 + r1)) over a scalar = 2 - A/B - C/D
        out[0] = 2.0f - (A / B) - (C / D);
    }
}

extern "C" void kernel_launch(void* const* d_in, const int* in_sizes, int n_in,
                              void* d_out, int out_size, void* d_ws, size_t ws_size,
                              hipStream_t stream) {
    (void)in_sizes; (void)n_in; (void)out_size; (void)ws_size;
    const float* img = (const float*)d_in[0];   // (1,1,30,30,30) fp32
    const float* enc = (const float*)d_in[1];   // (1,2,30,30,30) fp32; channel 0 used
    float* out = (float*)d_out;                 // scalar fp32 loss
    float* ws  = (float*)d_ws;                  // NBLOCKS*4 floats of partials

    softncut_partials<<<NBLOCKS, BLOCK_THREADS, 0, stream>>>(img, enc, ws);
    softncut_finalize<<<1, 32, 0, stream>>>(ws, out);
}